// GreenAmptInfiltration_10960756540276
// MI455X (gfx1250) — compile-verified
//
#include <hip/hip_runtime.h>
#include <cstdint>

// Green-Ampt infiltration scan: B=16384 rows x T=2048 steps, fp32.
// Memory-bound (536 MB traffic @ 23.3 TB/s ~ 23 us floor). Sequential in T,
// parallel in B (hard cap: 16384 threads = 512 wave32s). CDNA5 path:
// async global->LDS staging (ASYNCcnt, triple-buffered, 2 tiles of prefetch),
// transposed LDS tiles, coalesced NT b128 output stores for max store MLP.

#define BLOCK 128            // threads per block = rows per block (4 wave32s)
#define TC    16             // time-steps per tile
#define SPAD  (BLOCK + 2)    // padded transposed-tile row stride (floats);
                             // +2 => bank = (2*col + row) % 64, conflict-free

#define GA_EPS  1e-6f
#define GA_MINF 0.1f

typedef __attribute__((ext_vector_type(4))) float v4f;

__device__ __forceinline__ void issue_tile_loads_async(
    const float* __restrict__ precip, float* dstTile,
    int r0, int c0, int T, int tid)
{
  const uint64_t gbase = (uint64_t)(uintptr_t)precip;
#pragma unroll
  for (int j = 0; j < TC; ++j) {
    const int q  = tid + BLOCK * j;     // linear element id in 128x16 tile
    const int rr = q >> 4;              // row within tile
    const int cc = q & 15;              // col within tile (lane-contiguous)
    const uint32_t goff = (uint32_t)((r0 + rr) * T + (c0 + cc)) * 4u;
    // low 32 bits of a generic shared pointer == LDS byte offset
    const uint32_t lds = (uint32_t)(uintptr_t)(dstTile + cc * SPAD + rr);
    // CDNA5 async copy: global -> LDS, tracked by ASYNCcnt, no VGPR return
    asm volatile("global_load_async_to_lds_b32 %0, %1, %2"
                 :: "v"(lds), "v"(goff), "s"(gbase)
                 : "memory");
  }
}

__global__ __launch_bounds__(BLOCK)
void greenampt_scan_kernel(const float* __restrict__ precip,
                           const float* __restrict__ K,
                           const float* __restrict__ psi,
                           const float* __restrict__ dtheta,
                           float* __restrict__ out_infil,
                           float* __restrict__ out_runoff,
                           float* __restrict__ out_cumF,
                           int T)
{
  __shared__ float pbuf[3][TC * SPAD];   // precip tiles (transposed), 3-deep
  __shared__ float abuf[TC * SPAD];      // infil tile  (transposed)
  __shared__ float cbuf[TC * SPAD];      // cumF  tile  (transposed)

  const int tid = threadIdx.x;
  const int r0  = blockIdx.x * BLOCK;
  const int row = r0 + tid;

  // Per-row parameters: (B,1) arrays -> coalesced b32 loads
  const float Kv  = K[row];
  const float Kpd = Kv * (psi[row] * dtheta[row]);   // K * psi * dtheta
  float F = 0.0f;

  const int numTiles = T / TC;

  // Prologue: prefetch tiles 0 and 1 (16 async loads each per wave)
  issue_tile_loads_async(precip, pbuf[0], r0, 0, T, tid);
  if (numTiles > 1)
    issue_tile_loads_async(precip, pbuf[1], r0, TC, T, tid);

  for (int tile = 0; tile < numTiles; ++tile) {
    const int   c0 = tile * TC;
    float* pt = pbuf[tile % 3];

    // Retire exactly this tile's 16 async loads (they complete in order);
    // keep the next tile's 16 loads in flight for latency hiding.
    if (tile + 1 < numTiles)
      asm volatile("s_wait_asynccnt 16" ::: "memory");
    else
      asm volatile("s_wait_asynccnt 0" ::: "memory");
    __syncthreads();   // tile fully resident for all waves

    // Prefetch tile+2 into the buffer last read in the previous iteration.
    if (tile + 2 < numTiles)
      issue_tile_loads_async(precip, pbuf[(tile + 2) % 3], r0, c0 + 2 * TC,
                             T, tid);

    // Sequential Green-Ampt recurrence for this lane's row over TC steps.
    // v_rcp_f32 replaces the IEEE divide: fcap = Kv + Kpd / max(F, eps)
#pragma unroll
    for (int c = 0; c < TC; ++c) {
      const float P    = pt[c * SPAD + tid];               // conflict-free
      const float m    = fmaxf(F, GA_EPS);
      const float fcap = fmaxf(fmaf(Kpd, __builtin_amdgcn_rcpf(m), Kv), GA_MINF);
      const float fact = fminf(P, fcap);
      F += fact;                                           // DT = 1.0
      abuf[c * SPAD + tid] = fact;
      cbuf[c * SPAD + tid] = F;
    }
    __syncthreads();   // infil/cumF tiles complete for all waves

    // Coalesced write-out in float4 chunks: each thread owns 4 consecutive
    // columns of one row -> NT b128 stores (4x the per-wave in-flight store
    // bytes vs b32: 63 * 512B ~ 32KB/wave). Lanes: 4 per row, 8 rows/wave,
    // 64B contiguous per row segment. Transposed-LDS gathers stay
    // conflict-free (banks = 8*ccg + 2k + rr, rr 8-aligned per wave).
#pragma unroll
    for (int j = 0; j < TC / 4; ++j) {
      const int q   = tid + BLOCK * j;     // chunk id in [0, BLOCK*TC/4)
      const int rr  = q >> 2;              // row within tile
      const int cc  = (q & 3) * 4;         // starting col (multiple of 4)
      const int gi  = (r0 + rr) * T + (c0 + cc);

      v4f fa, Fv, ro;
#pragma unroll
      for (int k = 0; k < 4; ++k) {
        const float P  = pt  [(cc + k) * SPAD + rr];
        const float a  = abuf[(cc + k) * SPAD + rr];
        const float Cf = cbuf[(cc + k) * SPAD + rr];
        fa[k] = a;
        ro[k] = fmaxf(P - a, 0.0f);        // runoff = P - f_act (exact, >=0)
        Fv[k] = Cf;
      }
      __builtin_nontemporal_store(fa, (v4f*)(out_infil  + gi));
      __builtin_nontemporal_store(ro, (v4f*)(out_runoff + gi));
      __builtin_nontemporal_store(Fv, (v4f*)(out_cumF   + gi));
    }
    __syncthreads();   // done reading tiles before next iteration overwrites
  }
}

extern "C" void kernel_launch(void* const* d_in, const int* in_sizes, int n_in,
                              void* d_out, int out_size, void* d_ws, size_t ws_size,
                              hipStream_t stream)
{
  const float* precip = (const float*)d_in[0];   // (B, T)
  const float* K      = (const float*)d_in[1];   // (B, 1)
  const float* psi    = (const float*)d_in[2];   // (B, 1)
  const float* dth    = (const float*)d_in[3];   // (B, 1)

  const int B = in_sizes[1];                     // 16384
  const int T = in_sizes[0] / B;                 // 2048

  float* out    = (float*)d_out;                 // [infil | runoff | cumF]
  float* infil  = out;
  float* runoff = out + (size_t)B * T;
  float* cumF   = out + 2 * (size_t)B * T;

  const dim3 grid(B / BLOCK);   // 128 blocks of 128 threads (one thread/row)
  greenampt_scan_kernel<<<grid, BLOCK, 0, stream>>>(
      precip, K, psi, dth, infil, runoff, cumF, T);
}